// LIFNeuron_53145925320880
// MI455X (gfx1250) — compile-verified
//
#include <hip/hip_runtime.h>

// LIF neuron forward scan, T=64 steps over a (B*F)=262144-float plane.
// Memory-bound: 192 MB total traffic -> ~8.2 us at 23.3 TB/s.
// Strategy: one thread per float4 of the plane; 8-step-deep async
// global->LDS pipeline along T (ASYNCcnt), NT stores for the 128 MB output.

#define T_STEPS  64
#define PF_DEPTH 8              // async prefetch depth (steps in flight per wave)
#define NSLOTS   (2 * PF_DEPTH) // LDS ring slots (rewrite distance = 8 steps)
#define BLOCK    256

typedef float float4v __attribute__((ext_vector_type(4)));

// Builtin signature (from round-1 diagnostic): parameters are
// v4i addrspace(1)* and v4i addrspace(3)* with imm offset / imm cpol.
typedef int v4i __attribute__((vector_size(16)));
typedef __attribute__((address_space(1))) v4i gv4i;
typedef __attribute__((address_space(3))) v4i lv4i;

#if __has_builtin(__builtin_amdgcn_global_load_async_to_lds_b128)
#pragma message("probe: HAVE __builtin_amdgcn_global_load_async_to_lds_b128")
#endif
#if __has_builtin(__builtin_amdgcn_s_wait_asynccnt)
#pragma message("probe: HAVE __builtin_amdgcn_s_wait_asynccnt")
#endif

#if __has_builtin(__builtin_amdgcn_global_load_async_to_lds_b128) && \
    __has_builtin(__builtin_amdgcn_s_wait_asynccnt)
#define USE_ASYNC_LDS 1
#endif

// mem = 0.5*mem + x; spike = (mem >= 1); mem = spike ? 0 : mem;
// store spike and post-reset mem for this step (both NT: streamed once).
#define LIF_STEP(XV, T)                                                        \
  do {                                                                         \
    float4v m_ = mem * 0.5f + (XV);                                            \
    float4v sp_;                                                               \
    _Pragma("unroll") for (int k_ = 0; k_ < 4; ++k_) {                         \
      const bool fire_ = m_[k_] >= 1.0f;                                       \
      sp_[k_] = fire_ ? 1.0f : 0.0f;                                           \
      m_[k_]  = fire_ ? 0.0f : m_[k_];                                         \
    }                                                                          \
    mem = m_;                                                                  \
    __builtin_nontemporal_store(                                               \
        sp_, (float4v*)(spike + (size_t)(T) * plane + base));                  \
    __builtin_nontemporal_store(                                               \
        m_, (float4v*)(memseq + (size_t)(T) * plane + base));                  \
  } while (0)

__global__ __launch_bounds__(BLOCK) void lif_fwd(const float* __restrict__ x,
                                                 float* __restrict__ spike,
                                                 float* __restrict__ memseq,
                                                 int plane, int nvec) {
#ifdef USE_ASYNC_LDS
  __shared__ __align__(16) float lds_x[NSLOTS * BLOCK * 4]; // 64 KB ring
#endif
  const int gid = (int)(blockIdx.x * blockDim.x + threadIdx.x); // float4 index
  if (gid >= nvec) return;
  const size_t base = (size_t)gid * 4;

  float4v mem = {0.0f, 0.0f, 0.0f, 0.0f}; // V_REST = 0

#ifdef USE_ASYNC_LDS
  const int lbase = (int)threadIdx.x * 4; // float offset within a slot

  // Prologue: put PF_DEPTH steps in flight (per-wave ASYNCcnt = PF_DEPTH).
#pragma unroll
  for (int i = 0; i < PF_DEPTH; ++i) {
    __builtin_amdgcn_global_load_async_to_lds_b128(
        (gv4i*)(void*)(x + (size_t)i * plane + base),
        (lv4i*)(void*)(lds_x + (i % NSLOTS) * (BLOCK * 4) + lbase),
        /*offset=*/0, /*cpol=*/0);
  }

  // Steady state: retire oldest (async ops complete in order), consume from
  // LDS, compute+store, then refill one step 8 ahead into a slot whose last
  // reader finished 8 iterations ago (no DS/async WAR hazard).
#pragma unroll
  for (int t = 0; t < T_STEPS - PF_DEPTH; ++t) {
    __builtin_amdgcn_s_wait_asynccnt(PF_DEPTH - 1);
    float4v xv = *(const float4v*)(lds_x + (t % NSLOTS) * (BLOCK * 4) + lbase);
    LIF_STEP(xv, t);
    __builtin_amdgcn_global_load_async_to_lds_b128(
        (gv4i*)(void*)(x + (size_t)(t + PF_DEPTH) * plane + base),
        (lv4i*)(void*)(lds_x + ((t + PF_DEPTH) % NSLOTS) * (BLOCK * 4) + lbase),
        /*offset=*/0, /*cpol=*/0);
  }

  // Epilogue: everything is already in LDS.
  __builtin_amdgcn_s_wait_asynccnt(0);
#pragma unroll
  for (int t = T_STEPS - PF_DEPTH; t < T_STEPS; ++t) {
    float4v xv = *(const float4v*)(lds_x + (t % NSLOTS) * (BLOCK * 4) + lbase);
    LIF_STEP(xv, t);
  }
#else
  // Fallback: 8-deep register software pipeline (32 VGPRs of prefetch buffer).
  float4v buf[PF_DEPTH];
#pragma unroll
  for (int i = 0; i < PF_DEPTH; ++i)
    buf[i] = __builtin_nontemporal_load(
        (const float4v*)(x + (size_t)i * plane + base));
#pragma unroll
  for (int t = 0; t < T_STEPS; ++t) {
    float4v xv = buf[t % PF_DEPTH];
    if (t + PF_DEPTH < T_STEPS)
      buf[t % PF_DEPTH] = __builtin_nontemporal_load(
          (const float4v*)(x + (size_t)(t + PF_DEPTH) * plane + base));
    LIF_STEP(xv, t);
  }
#endif
}

extern "C" void kernel_launch(void* const* d_in, const int* in_sizes, int n_in,
                              void* d_out, int out_size, void* d_ws, size_t ws_size,
                              hipStream_t stream) {
  (void)n_in; (void)d_ws; (void)ws_size; (void)out_size;
  const float* x = (const float*)d_in[0];
  const int total = in_sizes[0];          // T * B * F = 16,777,216
  const int plane = total / T_STEPS;      // B * F     = 262,144
  const int nvec  = plane / 4;            // float4 lanes = 65,536
  float* spike  = (float*)d_out;          // outputs concatenated: spike_seq
  float* memseq = (float*)d_out + (size_t)total; // then mem_seq

  dim3 block(BLOCK);
  dim3 grid((nvec + BLOCK - 1) / BLOCK);  // 256 blocks
  hipLaunchKernelGGL(lif_fwd, grid, block, 0, stream, x, spike, memseq, plane, nvec);
}